// SpatialAttractorLoss_11115375362087
// MI455X (gfx1250) — compile-verified
//
#include <hip/hip_runtime.h>
#include <math.h>

typedef float v2f __attribute__((ext_vector_type(2)));
typedef float v8f __attribute__((ext_vector_type(8)));

#define B_ 16
#define C_ 11
#define H_ 64
#define W_ 64
#define HW_ (H_ * W_)
#define NTILE_ (B_ * HW_ / 16)   // 4096 query tiles total
#define TAU_ 1.5f

// ---------------- workspace layout (bytes) ----------------
#define WS_COUNTS 0
#define WS_OFFSETS 1024
#define WS_CURSORS 2048
#define WS_SORTED 4096
#define WS_PARTIAL (WS_SORTED + B_ * HW_ * 4)

__global__ void k_init(int* counts, float* partials) {
  int i = blockIdx.x * blockDim.x + threadIdx.x;
  if (i < B_ * C_) counts[i] = 0;
  if (i < NTILE_) partials[i] = 0.0f;
}

__global__ void k_count(const int* __restrict__ targets, int* __restrict__ counts) {
  int i = blockIdx.x * blockDim.x + threadIdx.x;
  if (i >= B_ * HW_) return;
  int t = targets[i];
  int b = i / HW_;
  if (t >= 0 && t < C_) atomicAdd(&counts[b * C_ + t], 1);
}

__global__ void k_scan(const int* __restrict__ counts, int* __restrict__ offsets,
                       int* __restrict__ cursors) {
  if (blockIdx.x == 0 && threadIdx.x == 0) {
    int acc = 0;
    for (int i = 0; i < B_ * C_; ++i) {
      offsets[i] = acc;
      cursors[i] = acc;
      acc += counts[i];
    }
  }
}

__global__ void k_scatter(const int* __restrict__ targets, int* __restrict__ cursors,
                          int* __restrict__ sorted) {
  int i = blockIdx.x * blockDim.x + threadIdx.x;
  if (i >= B_ * HW_) return;
  int t = targets[i];
  int b = i / HW_;
  int p = i - b * HW_;
  if (t >= 0 && t < C_) {
    int pos = atomicAdd(&cursors[b * C_ + t], 1);
    sorted[pos] = p;
  }
}

// Build the A-matrix operand (16 candidates x K=4) for one candidate tile.
//   A[m,:] = [my^2+mx^2, 1, -2*my, -2*mx]
__device__ __forceinline__ v2f make_a(const int* __restrict__ sorted, int base,
                                      int cnt, int j, int hi) {
  const int jj = (j < cnt) ? j : (cnt - 1);
  const int p = sorted[base + jj];
  float my = (float)(p >> 6);
  float mxx = (float)(p & 63);
  my = (j < cnt) ? my : 30000.0f;   // pad: far away, never wins min
  mxx = (j < cnt) ? mxx : 30000.0f;
  v2f a;
  const float s = my * my + mxx * mxx;
  a.x = hi ? (-2.0f * my) : s;
  a.y = hi ? (-2.0f * mxx) : 1.0f;
  return a;
}

// One wave (32 lanes) handles one tile of 16 query pixels.
// WMMA f32 16x16x4 computes a 16(candidates) x 16(queries) block of squared
// distances per instruction:  d^2 = A(m,:) . B(:,n)  with
//   B[:,n] = [1, qy^2+qx^2, qy, qx]
__global__ void __launch_bounds__(128)
k_main(const float* __restrict__ logits, const int* __restrict__ counts,
       const int* __restrict__ offsets, const int* __restrict__ sorted,
       float* __restrict__ partials) {
  const int lane = threadIdx.x & 31;
  const int wave = threadIdx.x >> 5;
  // Force wave-uniform values into SGPRs: keeps all loop control scalar so
  // EXEC is provably all-ones at every WMMA (ISA requirement).
  const int gtile =
      __builtin_amdgcn_readfirstlane(blockIdx.x * (blockDim.x >> 5) + wave);
  const int b = gtile >> 8;      // 256 tiles per batch image
  const int qtile = gtile & 255;
  const int n = lane & 15;       // query column within tile
  const int hi = lane >> 4;      // half-wave selects K pairs {0,1} vs {2,3}

  const int qpix = qtile * 16 + n;
  const float qy = (float)(qpix >> 6);
  const float qx = (float)(qpix & 63);

  v2f bop;  // B matrix 4x16, per lane: {K=hi?2:0, K=hi?3:1}
  bop.x = hi ? qy : 1.0f;
  bop.y = hi ? qx : (qy * qy + qx * qx);

  // ---- softmax over the 11 classes for this query pixel ----
  float lg[C_];
  const float* lp = logits + (size_t)b * C_ * HW_ + qpix;
  float mx = -3.0e38f;
#pragma unroll
  for (int c = 0; c < C_; ++c) {
    lg[c] = lp[(size_t)c * HW_];
    mx = fmaxf(mx, lg[c]);
  }
  float denom = 0.0f;
#pragma unroll
  for (int c = 0; c < C_; ++c) {
    lg[c] = __expf(lg[c] - mx);
    denom += lg[c];
  }
  const float inv = 1.0f / denom;

  float accum = 0.0f;  // sum_c probs * rewards (class 0 excluded: reward 0)
  for (int c = 1; c < C_; ++c) {
    const int cnt = __builtin_amdgcn_readfirstlane(counts[b * C_ + c]);
    if (cnt <= 0) continue;  // scalar branch; empty class -> reward 0
    const int base = __builtin_amdgcn_readfirstlane(offsets[b * C_ + c]);

    v8f minv;
#pragma unroll
    for (int r = 0; r < 8; ++r) minv[r] = 3.0e38f;

    // 2x unrolled: two independent back-to-back WMMAs per iteration.
    for (int t0 = 0; t0 < cnt; t0 += 32) {
      const v2f a0 = make_a(sorted, base, cnt, t0 + n, hi);
      const v2f a1 = make_a(sorted, base, cnt, t0 + 16 + n, hi);
      v8f zero = {};
      v8f d0 = __builtin_amdgcn_wmma_f32_16x16x4_f32(
          false, a0, false, bop, (short)0, zero, false, false);
      v8f d1 = __builtin_amdgcn_wmma_f32_16x16x4_f32(
          false, a1, false, bop, (short)0, zero, false, false);
#pragma unroll
      for (int r = 0; r < 8; ++r)
        minv[r] = fminf(minv[r], fminf(d0[r], d1[r]));
    }

    // D layout: lane (n) holds M=r (lanes 0-15) / M=r+8 (lanes 16-31).
    float mm = minv[0];
#pragma unroll
    for (int r = 1; r < 8; ++r) mm = fminf(mm, minv[r]);
    const float other = __shfl_xor(mm, 16, 32);  // combine M halves, same N
    const float d2 = fmaxf(fminf(mm, other), 0.0f);
    const float reward = __expf(-sqrtf(d2) * (1.0f / TAU_));
    accum += (lg[c] * inv) * reward;
  }

  // lanes 16..31 computed duplicates of lanes 0..15 -> mask them out.
  float v = (lane < 16) ? accum : 0.0f;
  for (int off = 16; off >= 1; off >>= 1) v += __shfl_xor(v, off, 32);
  if (lane == 0) partials[gtile] = v;
}

__global__ void k_reduce(const float* __restrict__ partials, float* __restrict__ out) {
  __shared__ float sm[256];
  float s = 0.0f;
  for (int i = threadIdx.x; i < NTILE_; i += 256) s += partials[i];
  sm[threadIdx.x] = s;
  __syncthreads();
  for (int st = 128; st > 0; st >>= 1) {
    if (threadIdx.x < st) sm[threadIdx.x] += sm[threadIdx.x + st];
    __syncthreads();
  }
  if (threadIdx.x == 0) out[0] = -sm[0] / (float)(B_ * HW_);
}

extern "C" void kernel_launch(void* const* d_in, const int* in_sizes, int n_in,
                              void* d_out, int out_size, void* d_ws, size_t ws_size,
                              hipStream_t stream) {
  const float* logits = (const float*)d_in[0];
  const int* targets = (const int*)d_in[1];
  float* out = (float*)d_out;

  char* ws = (char*)d_ws;
  int* counts = (int*)(ws + WS_COUNTS);
  int* offsets = (int*)(ws + WS_OFFSETS);
  int* cursors = (int*)(ws + WS_CURSORS);
  int* sorted = (int*)(ws + WS_SORTED);
  float* partials = (float*)(ws + WS_PARTIAL);

  k_init<<<(NTILE_ + 255) / 256, 256, 0, stream>>>(counts, partials);
  k_count<<<(B_ * HW_ + 255) / 256, 256, 0, stream>>>(targets, counts);
  k_scan<<<1, 32, 0, stream>>>(counts, offsets, cursors);
  k_scatter<<<(B_ * HW_ + 255) / 256, 256, 0, stream>>>(targets, cursors, sorted);
  // 4 waves per block, one 16-query tile per wave.
  k_main<<<NTILE_ / 4, 128, 0, stream>>>(logits, counts, offsets, sorted, partials);
  k_reduce<<<1, 256, 0, stream>>>(partials, out);
}